// Gat_layer_11012296146998
// MI455X (gfx1250) — compile-verified
//
#include <hip/hip_runtime.h>
#include <hip/hip_bf16.h>
#include <math.h>

#define NN    100000
#define HIDD  512
#define HEADS 8
#define FF    64
#define EE    800000

typedef __attribute__((ext_vector_type(2))) float v2f;
typedef __attribute__((ext_vector_type(8))) float v8f;

// ---------- helpers ----------
__device__ __forceinline__ int enc_f32(float f) {
    int i = __float_as_int(f);
    return i >= 0 ? i : (i ^ 0x7fffffff);  // monotone, self-inverse
}
__device__ __forceinline__ float dec_f32(int j) {
    return __int_as_float(j >= 0 ? j : (j ^ 0x7fffffff));
}

// ---------- K0: zero / init ----------
__global__ void k_zero(float* p, int n) {
    for (int i = blockIdx.x * blockDim.x + threadIdx.x; i < n; i += gridDim.x * blockDim.x)
        p[i] = 0.0f;
}
__global__ void k_init_gmax(int* gmax) {
    if (threadIdx.x == 0 && blockIdx.x == 0) *gmax = (int)0x807fffff;  // enc(-inf)
}

// ---------- K1: fp32 WMMA GEMM  proj = X * W^T + b_lin ----------
// block = 256 threads (8 waves). Block owns 16 rows of X; wave w owns cols [w*64, w*64+64).
#define XS_STRIDE 516  // 512 + 4 pad -> A-fragment LDS reads spread over 16 banks
__global__ void k_gemm(const float* __restrict__ X, const float* __restrict__ W,
                       const float* __restrict__ b_lin, float* __restrict__ proj) {
    __shared__ float xs[16 * XS_STRIDE];

    const int n0 = blockIdx.x * 16;

    // stage X strip [16 x 512] into LDS with float4 loads
    const float4* X4 = (const float4*)X;
#pragma unroll
    for (int j = 0; j < 8; ++j) {
        int v   = threadIdx.x + 256 * j;      // 0..2047 float4 slots
        int row = v >> 7;                     // 128 float4 per row
        int c4  = v & 127;
        float4 d = X4[(size_t)(n0 + row) * 128 + c4];
        *(float4*)&xs[row * XS_STRIDE + c4 * 4] = d;
    }
    __syncthreads();

    const int lane  = threadIdx.x & 31;
    const int wave  = threadIdx.x >> 5;
    const int mrow  = lane & 15;              // M (A) / N-col (B rows of W)
    const int khalf = (lane >> 4) << 1;       // 0 or 2
    const int h0    = wave * 64;

    v8f acc0 = {}, acc1 = {}, acc2 = {}, acc3 = {};

    const float2* w0 = (const float2*)(W + (size_t)(h0 +  0 + mrow) * HIDD);
    const float2* w1 = (const float2*)(W + (size_t)(h0 + 16 + mrow) * HIDD);
    const float2* w2 = (const float2*)(W + (size_t)(h0 + 32 + mrow) * HIDD);
    const float2* w3 = (const float2*)(W + (size_t)(h0 + 48 + mrow) * HIDD);

    for (int k = 0; k < HIDD; k += 4) {
        // A fragment from LDS (b64): lane m holds X[n0+m][k+khalf .. +1]
        const float2 af = *(const float2*)&xs[mrow * XS_STRIDE + k + khalf];
        v2f a; a.x = af.x; a.y = af.y;

        const int kq = (k + khalf) >> 1;
        float2 bf;
        v2f b;
        bf = w0[kq]; b.x = bf.x; b.y = bf.y;
        acc0 = __builtin_amdgcn_wmma_f32_16x16x4_f32(false, a, false, b, (short)0, acc0, false, false);
        bf = w1[kq]; b.x = bf.x; b.y = bf.y;
        acc1 = __builtin_amdgcn_wmma_f32_16x16x4_f32(false, a, false, b, (short)0, acc1, false, false);
        bf = w2[kq]; b.x = bf.x; b.y = bf.y;
        acc2 = __builtin_amdgcn_wmma_f32_16x16x4_f32(false, a, false, b, (short)0, acc2, false, false);
        bf = w3[kq]; b.x = bf.x; b.y = bf.y;
        acc3 = __builtin_amdgcn_wmma_f32_16x16x4_f32(false, a, false, b, (short)0, acc3, false, false);
    }

    // writeback: D vgpr j -> M = 8*(lane/16)+j, N = lane%16
    const int ncol  = lane & 15;
    const int mbase = (lane >> 4) * 8;
#pragma unroll
    for (int j = 0; j < 8; ++j) {
        size_t rbase = (size_t)(n0 + mbase + j) * HIDD + h0;
        proj[rbase +  0 + ncol] = acc0[j] + b_lin[h0 +  0 + ncol];
        proj[rbase + 16 + ncol] = acc1[j] + b_lin[h0 + 16 + ncol];
        proj[rbase + 32 + ncol] = acc2[j] + b_lin[h0 + 32 + ncol];
        proj[rbase + 48 + ncol] = acc3[j] + b_lin[h0 + 48 + ncol];
    }
}

// ---------- K2: per-node attention scores ----------
__global__ void k_scores(const float* __restrict__ proj, const float* __restrict__ a_src,
                         const float* __restrict__ a_tgt, float* __restrict__ ssrc,
                         float* __restrict__ stgt) {
    int idx = blockIdx.x * blockDim.x + threadIdx.x;
    if (idx >= NN * HEADS) return;
    int n = idx >> 3, h = idx & 7;
    const float4* p  = (const float4*)(proj + (size_t)n * HIDD + h * FF);
    const float4* as = (const float4*)(a_src + h * FF);
    const float4* at = (const float4*)(a_tgt + h * FF);
    float s = 0.f, t = 0.f;
#pragma unroll
    for (int i = 0; i < FF / 4; ++i) {
        float4 d = p[i], u = as[i], v = at[i];
        s += d.x * u.x + d.y * u.y + d.z * u.z + d.w * u.w;
        t += d.x * v.x + d.y * v.y + d.z * v.z + d.w * v.w;
    }
    ssrc[idx] = s;
    stgt[idx] = t;
}

// ---------- K3: global max over leaky-relu edge scores ----------
__global__ void k_edge_max(const int* __restrict__ ei, const float* __restrict__ ssrc,
                           const float* __restrict__ stgt, int* __restrict__ gmax) {
    float vmax = -INFINITY;
    const int total = EE * HEADS;
    for (int idx = blockIdx.x * blockDim.x + threadIdx.x; idx < total; idx += gridDim.x * blockDim.x) {
        int e = idx >> 3, h = idx & 7;
        int s = ei[e], t = ei[EE + e];
        float sc = ssrc[s * HEADS + h] + stgt[t * HEADS + h];
        sc = sc > 0.f ? sc : 0.01f * sc;
        vmax = fmaxf(vmax, sc);
    }
#pragma unroll
    for (int off = 16; off; off >>= 1) vmax = fmaxf(vmax, __shfl_xor(vmax, off, 32));
    if ((threadIdx.x & 31) == 0) atomicMax(gmax, enc_f32(vmax));
}

// ---------- K4: softmax denominator (segment sum over target) ----------
__global__ void k_denom(const int* __restrict__ ei, const float* __restrict__ ssrc,
                        const float* __restrict__ stgt, const int* __restrict__ gmax,
                        float* __restrict__ denom) {
    const float m = dec_f32(*gmax);
    const int total = EE * HEADS;
    for (int idx = blockIdx.x * blockDim.x + threadIdx.x; idx < total; idx += gridDim.x * blockDim.x) {
        int e = idx >> 3, h = idx & 7;
        int s = ei[e], t = ei[EE + e];
        float sc = ssrc[s * HEADS + h] + stgt[t * HEADS + h];
        sc = sc > 0.f ? sc : 0.01f * sc;
        atomicAdd(&denom[t * HEADS + h], expf(sc - m));
    }
}

// ---------- K5: weighted scatter aggregation (one wave per edge) ----------
__global__ void k_aggregate(const int* __restrict__ ei, const float* __restrict__ ssrc,
                            const float* __restrict__ stgt, const int* __restrict__ gmax,
                            const float* __restrict__ denom, const float* __restrict__ proj,
                            float* __restrict__ out) {
    int gwave = (blockIdx.x * blockDim.x + threadIdx.x) >> 5;
    if (gwave >= EE) return;
    int lane = threadIdx.x & 31;
    int s = ei[gwave], t = ei[EE + gwave];

    float attn = 0.f;
    if (lane < HEADS) {
        float sc = ssrc[s * HEADS + lane] + stgt[t * HEADS + lane];
        sc = sc > 0.f ? sc : 0.01f * sc;
        attn = expf(sc - dec_f32(*gmax)) / (denom[t * HEADS + lane] + 1e-16f);
    }

    const float4* ps = (const float4*)(proj + (size_t)s * HIDD);
    float* ob = out + (size_t)t * HIDD;
#pragma unroll
    for (int i = 0; i < 4; ++i) {
        int v4 = lane + 32 * i;            // float4 index 0..127
        int h  = v4 >> 4;                  // 16 float4 per head
        float a = __shfl(attn, h, 32);
        float4 d = ps[v4];
        float* o = ob + v4 * 4;
        atomicAdd(o + 0, d.x * a);
        atomicAdd(o + 1, d.y * a);
        atomicAdd(o + 2, d.z * a);
        atomicAdd(o + 3, d.w * a);
    }
}

// ---------- K6: bias + PReLU epilogue (in place) ----------
__global__ void k_finalize(float* __restrict__ out, const float* __restrict__ bias,
                           const float* __restrict__ prelu_a) {
    const float pa = *prelu_a;
    const int total = NN * HIDD;
    for (int i = blockIdx.x * blockDim.x + threadIdx.x; i < total; i += gridDim.x * blockDim.x) {
        float v = out[i] + bias[i & (HIDD - 1)];
        out[i] = v >= 0.f ? v : pa * v;
    }
}

// ---------- launch ----------
extern "C" void kernel_launch(void* const* d_in, const int* in_sizes, int n_in,
                              void* d_out, int out_size, void* d_ws, size_t ws_size,
                              hipStream_t stream) {
    const float* X      = (const float*)d_in[0];
    const int*   ei     = (const int*)d_in[1];
    const float* W      = (const float*)d_in[2];
    const float* b_lin  = (const float*)d_in[3];
    const float* a_src  = (const float*)d_in[4];
    const float* a_tgt  = (const float*)d_in[5];
    const float* bias   = (const float*)d_in[6];
    const float* prelua = (const float*)d_in[7];
    float* out = (float*)d_out;

    // workspace layout
    float* proj  = (float*)d_ws;                       // NN*HIDD
    float* ssrc  = proj + (size_t)NN * HIDD;           // NN*HEADS
    float* stgt  = ssrc + (size_t)NN * HEADS;          // NN*HEADS
    float* denom = stgt + (size_t)NN * HEADS;          // NN*HEADS
    int*   gmax  = (int*)(denom + (size_t)NN * HEADS); // 1

    k_zero<<<2048, 256, 0, stream>>>(out, NN * HIDD);
    k_zero<<<512, 256, 0, stream>>>(denom, NN * HEADS);
    k_init_gmax<<<1, 32, 0, stream>>>(gmax);

    k_gemm<<<NN / 16, 256, 0, stream>>>(X, W, b_lin, proj);

    k_scores<<<(NN * HEADS + 255) / 256, 256, 0, stream>>>(proj, a_src, a_tgt, ssrc, stgt);

    k_edge_max<<<4096, 256, 0, stream>>>(ei, ssrc, stgt, gmax);
    k_denom<<<8192, 256, 0, stream>>>(ei, ssrc, stgt, gmax, denom);

    k_aggregate<<<(EE * 32) / 256, 256, 0, stream>>>(ei, ssrc, stgt, gmax, denom, proj, out);

    k_finalize<<<4096, 256, 0, stream>>>(out, bias, prelua);
}